// LightGCN_36086315221009
// MI455X (gfx1250) — compile-verified
//
#include <hip/hip_runtime.h>

typedef __attribute__((ext_vector_type(2))) float v2f;
typedef __attribute__((ext_vector_type(8))) float v8f;

#define DIM 64

// ---------------- CSR row pointer via binary search (rows are sorted) -------
__global__ void k_rowptr(const int* __restrict__ rows, int* __restrict__ row_start,
                         int n_rows, int nnz) {
    int r = blockIdx.x * blockDim.x + threadIdx.x;
    if (r > n_rows) return;
    int lo = 0, hi = nnz;                 // lower_bound(rows, r)
    while (lo < hi) {
        int mid = (lo + hi) >> 1;
        if (rows[mid] < r) lo = mid + 1; else hi = mid;
    }
    row_start[r] = lo;
}

__global__ void k_copy(const float* __restrict__ a, float* __restrict__ o, int n) {
    int i = blockIdx.x * blockDim.x + threadIdx.x;
    if (i < n) o[i] = a[i];
}

__global__ void k_scale(float* __restrict__ o, float s, int n) {
    int i = blockIdx.x * blockDim.x + threadIdx.x;
    if (i < n) o[i] *= s;
}

// ---------------- per-row Pearson(x_row, y_row), one wave32 per row ---------
__global__ void k_pearson(const float* __restrict__ x, const float* __restrict__ y,
                          float* __restrict__ bout, int n_rows) {
    int wave = (blockIdx.x * blockDim.x + threadIdx.x) >> 5;
    int lane = threadIdx.x & 31;
    if (wave >= n_rows) return;
    const float* xr = x + (size_t)wave * DIM;
    const float* yr = y + (size_t)wave * DIM;
    float x0 = xr[2 * lane], x1 = xr[2 * lane + 1];
    float y0 = yr[2 * lane], y1 = yr[2 * lane + 1];
    float sx = x0 + x1, sy = y0 + y1;
    for (int m = 16; m >= 1; m >>= 1) {
        sx += __shfl_xor(sx, m, 32);
        sy += __shfl_xor(sy, m, 32);
    }
    float mx = sx * (1.0f / 64.0f), my = sy * (1.0f / 64.0f);
    float xd0 = x0 - mx, xd1 = x1 - mx, yd0 = y0 - my, yd1 = y1 - my;
    float num = xd0 * yd0 + xd1 * yd1;
    float sxx = xd0 * xd0 + xd1 * xd1;
    float syy = yd0 * yd0 + yd1 * yd1;
    for (int m = 16; m >= 1; m >>= 1) {
        num += __shfl_xor(num, m, 32);
        sxx += __shfl_xor(sxx, m, 32);
        syy += __shfl_xor(syy, m, 32);
    }
    float den = sqrtf(sxx) * sqrtf(syy);
    den = (den < 1e-7f) ? 1e-7f : den;
    if (lane == 0) bout[wave] = num / den;
}

// ---------------- segmented SpMM via V_WMMA_F32_16X16X4_F32 -----------------
// One wave32 handles 16 consecutive output rows (a contiguous edge range,
// since rows are sorted). D(16x64) = A(16xK) * B(Kx64) accumulated in K-steps
// of 4 edges; A[m][k] = vals[k] iff rows[k] == rowBase+m, B[k][:] = X[cols[k]].
// Edge metadata is coalesced-loaded once per 32-edge chunk (one edge per lane)
// and broadcast per K-step via ds_bpermute (__shfl) -- no per-step VMEM for
// metadata, and the inner body is branch-free (tail handled by cndmask only).
// Fragment layouts per CDNA5 ISA 7.12.2 (f32, 16x16x4):
//   A/B: VGPR0 -> K = 0 (lanes 0-15) / K = 2 (lanes 16-31); VGPR1 -> K = 1 / 3
//   D:   VGPR p -> row p (lanes 0-15) / row p+8 (lanes 16-31), col = lane&15
template <bool BLEND, bool WRITE_NEXT>
__global__ __launch_bounds__(256)
void k_spmm_wmma(const float* __restrict__ X,      // source embeddings (gathered)
                 const float* __restrict__ E0,     // base embeddings (blend)
                 const float* __restrict__ Bcoef,  // per-row pearson coefficient
                 const int* __restrict__ rows,
                 const int* __restrict__ cols,
                 const float* __restrict__ vals,
                 const int* __restrict__ row_start,
                 float* __restrict__ Enext,        // this layer's embedding
                 float* __restrict__ Osum,         // running sum for the mean
                 int n_rows) {
    const int wavesPerBlock = blockDim.x >> 5;
    int waveId = blockIdx.x * wavesPerBlock + (threadIdx.x >> 5);
    int lane = threadIdx.x & 31;
    int rowBase = waveId * 16;
    if (rowBase >= n_rows) return;                 // whole-wave uniform exit
    int rbEnd = rowBase + 16; if (rbEnd > n_rows) rbEnd = n_rows;
    int k0 = row_start[rowBase];
    int kend = row_start[rbEnd];

    v8f acc0 = {}, acc1 = {}, acc2 = {}, acc3 = {};
    const int m      = lane & 15;        // row within group (A) / column (B,D)
    const int kk     = (lane >> 4) << 1; // K pair: lanes 0-15 -> {0,1}, 16-31 -> {2,3}
    const int target = rowBase + m;

    for (int kbase = k0; kbase < kend; kbase += 32) {
        // coalesced metadata load: one edge per lane (clamped => always in-bounds)
        int ke  = kbase + lane;
        int keC = (ke < kend) ? ke : (kend - 1);
        int   r32 = rows[keC];
        int   c32 = cols[keC];
        float w32 = vals[keC];
#pragma unroll
        for (int j = 0; j < 8; ++j) {
            if (kbase + 4 * j >= kend) break;      // wave-uniform: no divergence
            int idxA = 4 * j + kk;                 // edge slot within chunk
            int idxB = idxA + 1;
            int   rA = __shfl(r32, idxA, 32), rB = __shfl(r32, idxB, 32);
            int   cA = __shfl(c32, idxA, 32), cB = __shfl(c32, idxB, 32);
            float wA = __shfl(w32, idxA, 32), wB = __shfl(w32, idxB, 32);
            bool va = (kbase + idxA) < kend;       // lane-half select, cndmask only
            bool vb = (kbase + idxB) < kend;
            v2f a;
            a.x = (va && rA == target) ? wA : 0.0f;
            a.y = (vb && rB == target) ? wB : 0.0f;
            const float* xA = X + (size_t)cA * DIM;  // stale cols killed by A==0
            const float* xB = X + (size_t)cB * DIM;
            v2f b0, b1, b2, b3;
            b0.x = xA[m];      b0.y = xB[m];
            b1.x = xA[16 + m]; b1.y = xB[16 + m];
            b2.x = xA[32 + m]; b2.y = xB[32 + m];
            b3.x = xA[48 + m]; b3.y = xB[48 + m];
            acc0 = __builtin_amdgcn_wmma_f32_16x16x4_f32(false, a, false, b0, (short)0, acc0, false, false);
            acc1 = __builtin_amdgcn_wmma_f32_16x16x4_f32(false, a, false, b1, (short)0, acc1, false, false);
            acc2 = __builtin_amdgcn_wmma_f32_16x16x4_f32(false, a, false, b2, (short)0, acc2, false, false);
            acc3 = __builtin_amdgcn_wmma_f32_16x16x4_f32(false, a, false, b3, (short)0, acc3, false, false);
        }
    }

    // Epilogue: component p of tile t -> row rowBase + p + 8*(lane>=16), col 16*t + (lane&15)
    int rOff = rowBase + (lane >> 4) * 8;
#pragma unroll
    for (int p = 0; p < 8; ++p) {
        int r = rOff + p;
        if (r >= n_rows) continue;
        size_t base = (size_t)r * DIM + m;
        float d0 = acc0[p], d1 = acc1[p], d2 = acc2[p], d3 = acc3[p];
        if (BLEND) {
            float bc = Bcoef[r], ob = 1.0f - bc;
            d0 = bc * d0 + ob * E0[base];
            d1 = bc * d1 + ob * E0[base + 16];
            d2 = bc * d2 + ob * E0[base + 32];
            d3 = bc * d3 + ob * E0[base + 48];
        }
        if (WRITE_NEXT) {
            Enext[base]      = d0;
            Enext[base + 16] = d1;
            Enext[base + 32] = d2;
            Enext[base + 48] = d3;
        }
        Osum[base]      += d0;   // single-writer per row: no atomics needed
        Osum[base + 16] += d1;
        Osum[base + 32] += d2;
        Osum[base + 48] += d3;
    }
}

extern "C" void kernel_launch(void* const* d_in, const int* in_sizes, int n_in,
                              void* d_out, int out_size, void* d_ws, size_t ws_size,
                              hipStream_t stream) {
    const float* e0    = (const float*)d_in[0];
    const float* vals  = (const float*)d_in[1];
    const float* vals2 = (const float*)d_in[2];
    const int*   rows  = (const int*)d_in[3];
    const int*   cols  = (const int*)d_in[4];
    const int*   rows2 = (const int*)d_in[5];
    const int*   cols2 = (const int*)d_in[6];
    float* out = (float*)d_out;

    const int N    = in_sizes[0] / DIM;
    const int NNZ1 = in_sizes[3];
    const int NNZ2 = in_sizes[5];
    const int NE   = N * DIM;

    // workspace carve-up
    float* eA    = (float*)d_ws;              // N*DIM
    float* eB    = eA + (size_t)NE;           // N*DIM
    float* bcoef = eB + (size_t)NE;           // N
    int*   rs1   = (int*)(bcoef + N);         // N+1
    int*   rs2   = rs1 + (N + 1);             // N+1

    const int T = 256;
    dim3 blk(T);

    k_rowptr<<<dim3((N + 1 + T - 1) / T), blk, 0, stream>>>(rows, rs1, N, NNZ1);
    k_rowptr<<<dim3((N + 1 + T - 1) / T), blk, 0, stream>>>(rows2, rs2, N, NNZ2);

    // Osum starts as e0 (embs2[0])
    k_copy<<<dim3((NE + T - 1) / T), blk, 0, stream>>>(e0, out, NE);

    const int groups = (N + 15) / 16;
    const int wpb = T / 32;                        // 8 waves (16-row groups) per block
    dim3 spmmGrid((groups + wpb - 1) / wpb);
    dim3 pearGrid((N + wpb - 1) / wpb);

    // embs2[1] = spmm(graph1, e0), no blend
    k_spmm_wmma<false, true><<<spmmGrid, blk, 0, stream>>>(
        e0, nullptr, nullptr, rows, cols, vals, rs1, eA, out, N);

    // 3 propagation layers on graph2 with pearson blend (ping-pong eA/eB);
    // the final layer's embedding is only needed inside Osum -> skip its store.
    float* cur = eA;
    float* nxt = eB;
    for (int layer = 0; layer < 3; ++layer) {
        k_pearson<<<pearGrid, blk, 0, stream>>>(cur, e0, bcoef, N);
        if (layer < 2) {
            k_spmm_wmma<true, true><<<spmmGrid, blk, 0, stream>>>(
                cur, e0, bcoef, rows2, cols2, vals2, rs2, nxt, out, N);
        } else {
            k_spmm_wmma<true, false><<<spmmGrid, blk, 0, stream>>>(
                cur, e0, bcoef, rows2, cols2, vals2, rs2, nxt, out, N);
        }
        float* t = cur; cur = nxt; nxt = t;
    }

    // mean over 5 embeddings; d_out == [users; items] == light_out flat
    k_scale<<<dim3((NE + T - 1) / T), blk, 0, stream>>>(out, 0.2f, NE);
}